// BACENET_17583596110324
// MI455X (gfx1250) — compile-verified
//
#include <hip/hip_runtime.h>
#include <hip/hip_bf16.h>

// ---------------------------------------------------------------------------
// BACENET angular-descriptor kernel for gfx1250 (MI455X, wave32, WMMA)
//
//   Phase 0: zero g workspace [nat][16][36] f32 (stride 36 = K padded to 9x4)
//   Phase 1: pair-parallel scatter: g[a][r][l] += radial[p][r] * prod u^l
//            (global_atomic_add_f32, L2-resident accumulator; ~136M atomics
//            dominate — problem is scatter-bound, ~50MB total traffic)
//   Phase 2: per-atom GEMM  out[n][r][z] = sum_l g^2[n][r][l] * lam[z]^s(l) * norm(l)
//            mapped onto v_wmma_f32_16x16x4_f32, K=36 in 9 chunks, one wave/atom.
//            All B-matrix indices compile-time: no arrays, no LDS/scratch.
// ---------------------------------------------------------------------------

#define NRAD   16
#define NLAM   4
#define NL     34
#define LSTRIDE 36   // NL padded to multiple of 4 (zero-filled)

typedef __attribute__((ext_vector_type(2))) float v2f;
typedef __attribute__((ext_vector_type(8))) float v8f;

// (lx,ly,lz) combos in exact reference order: s=1..4, lx ascending, ly ascending
__constant__ const int cLX[NL] = {0,0,1, 0,0,0,1,1,2, 0,0,0,0,1,1,1,2,2,3, 0,0,0,0,0,1,1,1,1,2,2,2,3,3,4};
__constant__ const int cLY[NL] = {0,1,0, 0,1,2,0,1,0, 0,1,2,3,0,1,2,0,1,0, 0,1,2,3,4,0,1,2,3,0,1,2,0,1,0};
__constant__ const int cLZ[NL] = {1,0,0, 2,1,0,1,0,0, 3,2,1,0,2,1,0,1,0,0, 4,3,2,1,0,3,2,1,0,2,1,0,1,0,0};

// s(l) = lx+ly+lz: compile-time function of monomial index (pad -> 0)
__host__ __device__ constexpr int sOf(int l) {
    return (l < 3) ? 1 : (l < 9) ? 2 : (l < 19) ? 3 : (l < NL) ? 4 : 0;
}

// ---------------------------------------------------------------------------
// Phase 1: one thread per (pair, radial channel). 34 atomic f32 adds each.
// ---------------------------------------------------------------------------
__global__ void bace_scatter(const float* __restrict__ rij_unit,   // [P,3]
                             const float* __restrict__ radial,     // [P,NRAD]
                             const int*   __restrict__ atom_idx,   // [P]
                             float*       __restrict__ g,          // [nat,NRAD,LSTRIDE]
                             int npairs)
{
    int t = blockIdx.x * blockDim.x + threadIdx.x;
    if (t >= npairs * NRAD) return;
    int p = t >> 4;
    int r = t & (NRAD - 1);

    float ux = rij_unit[3 * p + 0] + 1e-12f;
    float uy = rij_unit[3 * p + 1] + 1e-12f;
    float uz = rij_unit[3 * p + 2] + 1e-12f;
    float rv = radial[(size_t)p * NRAD + r];
    int   a  = atom_idx[p];

    // power tables; all indexing below is compile-time constant after unroll
    float px[5], py[5], pz[5];
    px[0] = 1.0f; py[0] = 1.0f; pz[0] = 1.0f;
    #pragma unroll
    for (int k = 1; k < 5; ++k) {
        px[k] = px[k - 1] * ux;
        py[k] = py[k - 1] * uy;
        pz[k] = pz[k - 1] * uz;
    }

    float* base = g + ((size_t)a * NRAD + r) * LSTRIDE;
    #pragma unroll
    for (int l = 0; l < NL; ++l) {
        float m = px[cLX[l]] * py[cLY[l]] * pz[cLZ[l]];
        atomicAdd(base + l, rv * m);   // global_atomic_add_f32, no return
    }
}

// ---------------------------------------------------------------------------
// Phase 2: one wave32 per atom. 16x16xK GEMM with K=36 via 9 WMMA f32 16x16x4.
//   A (16x4): M = radial channel, K = monomial chunk.  lanes 0-15: K={0,1},
//             lanes 16-31: K={2,3} (ISA 7.12.2 layout), values = g^2.
//   B (4x16): N = lambda index (cols 0..3 live, cols 4..15 zero via lam=0),
//             b[k][z] = lambda[z]^s(l) * fact_norm[l]  (s, indices compile-time).
//   C (16x16 f32): VGPR j holds M = j + 8*(lane>=16), N = lane%16.
// ---------------------------------------------------------------------------
__global__ void bace_contract(const float* __restrict__ g,        // [nat,NRAD,LSTRIDE]
                              const float* __restrict__ lam_w,    // [NLAM]
                              const float* __restrict__ fnorm,    // [NL]
                              const int*   __restrict__ zeta_p,   // [1]
                              float*       __restrict__ out,      // [nat,NRAD,NLAM]
                              int nat)
{
    int lane = threadIdx.x & 31;
    int wave = threadIdx.x >> 5;
    int n = blockIdx.x * (blockDim.x >> 5) + wave;
    if (n >= nat) return;                       // wave-uniform exit: EXEC stays all-1s

    int  col = lane & 15;                       // A: M row (radial) / B: N col (lambda)
    bool hi  = (lane >> 4) != 0;                // lanes 16-31 carry K={2,3} of each chunk
    float lamv = (col < NLAM) ? lam_w[col] : 0.0f;   // zero kills cols >= NLAM (all s>=1)

    // lamv^s for s = 0..4 (3 multiplies; all indexing compile-time below)
    float lampow[5];
    lampow[0] = 1.0f;
    lampow[1] = lamv;
    lampow[2] = lamv * lamv;
    lampow[3] = lampow[2] * lamv;
    lampow[4] = lampow[2] * lampow[2];

    // fact_norm with zero padding; constant indices, uniform scalar loads
    float fn[LSTRIDE];
    #pragma unroll
    for (int l = 0; l < NL; ++l) fn[l] = fnorm[l];
    fn[34] = 0.0f;
    fn[35] = 0.0f;

    const float* arow = g + ((size_t)n * NRAD + col) * LSTRIDE + (hi ? 2 : 0);

    v8f acc = {};
    #pragma unroll
    for (int c = 0; c < 9; ++c) {
        // A chunk: contiguous float2 of g, squared in-register
        v2f gv = *(const v2f*)(arow + c * 4);
        v2f a;
        a.x = gv.x * gv.x;
        a.y = gv.y * gv.y;

        // B chunk: lanes 0-15 take K={4c,4c+1}, lanes 16-31 take K={4c+2,4c+3}.
        // Both alternatives have compile-time l => compile-time s(l), fn index;
        // the only runtime ops are v_cndmask + v_mul.
        v2f b;
        b.x = hi ? lampow[sOf(4 * c + 2)] * fn[4 * c + 2]
                 : lampow[sOf(4 * c + 0)] * fn[4 * c + 0];
        b.y = hi ? lampow[sOf(4 * c + 3)] * fn[4 * c + 3]
                 : lampow[sOf(4 * c + 1)] * fn[4 * c + 1];

        // D = A x B + C  (v_wmma_f32_16x16x4_f32)
        acc = __builtin_amdgcn_wmma_f32_16x16x4_f32(
            /*neg_a=*/false, a, /*neg_b=*/false, b,
            /*c_mod=*/(short)0, acc, /*reuse_a=*/false, /*reuse_b=*/false);
    }

    float scale = exp2f(1.0f - (float)(*zeta_p));   // 2^(1-zeta)
    if (col < NLAM) {
        int rbase = hi ? 8 : 0;
        #pragma unroll
        for (int j = 0; j < 8; ++j) {
            out[((size_t)n * NRAD + (rbase + j)) * NLAM + col] = acc[j] * scale;
        }
    }
}

// ---------------------------------------------------------------------------
// Launch
// ---------------------------------------------------------------------------
extern "C" void kernel_launch(void* const* d_in, const int* in_sizes, int n_in,
                              void* d_out, int out_size, void* d_ws, size_t ws_size,
                              hipStream_t stream)
{
    // setup_inputs() order:
    // 0:z 1:r_idx 2:rij_unit 3:radial_ij 4:first_atom_idx 5:lambda_weights
    // 6:lxlylz 7:lxlylz_sum 8:fact_norm 9:nat
    const int*   zptr   = (const int*)  d_in[0];
    const float* rij    = (const float*)d_in[2];
    const float* radial = (const float*)d_in[3];
    const int*   idx    = (const int*)  d_in[4];
    const float* lamw   = (const float*)d_in[5];
    const float* fnorm  = (const float*)d_in[8];
    float*       out    = (float*)d_out;

    int npairs = in_sizes[4];
    int nat    = out_size / (NRAD * NLAM);    // 800000 / 64 = 12500

    float* g = (float*)d_ws;
    size_t gbytes = (size_t)nat * NRAD * LSTRIDE * sizeof(float);
    hipMemsetAsync(d_ws, 0, gbytes, stream);  // graph-capturable

    int total1 = npairs * NRAD;
    bace_scatter<<<(total1 + 255) / 256, 256, 0, stream>>>(rij, radial, idx, g, npairs);

    const int wavesPerBlock = 8;              // 256 threads = 8 wave32
    int grid2 = (nat + wavesPerBlock - 1) / wavesPerBlock;
    bace_contract<<<grid2, 32 * wavesPerBlock, 0, stream>>>(g, lamw, fnorm,
                                                            zptr, out, nat);
}